// Encoder_7318624272620
// MI455X (gfx1250) — compile-verified
//
#include <hip/hip_runtime.h>

typedef __attribute__((ext_vector_type(2))) float v2f;
typedef __attribute__((ext_vector_type(8))) float v8f;

#define N_NODES 50000
#define N_EDGES 800000
#define D 128
#define ROW_TILES (N_NODES / 16)   // 3125, exact

// ---------------- utility kernels ----------------

__global__ void zero_f32(float* __restrict__ p, long n) {
  long i = (long)blockIdx.x * blockDim.x + threadIdx.x;
  long stride = (long)gridDim.x * blockDim.x;
  for (; i < n; i += stride) p[i] = 0.0f;
}

__global__ void deg_count(const int* __restrict__ dst, float* __restrict__ deg, int n) {
  int e = blockIdx.x * blockDim.x + threadIdx.x;
  if (e < n) atomicAdd(&deg[dst[e]], 1.0f);
}

__global__ void finalize_inv_deg(float* __restrict__ deg, int n) {
  int i = blockIdx.x * blockDim.x + threadIdx.x;
  if (i < n) {
    float d = deg[i];
    deg[i] = (d > 0.0f) ? (1.0f / d) : 0.0f;
  }
}

// One wave per edge: gather one 128-float message row (float4/lane, 32 lanes)
// and atomically accumulate into agg[dst]. Gathers hit L2 (msg buf = 25.6MB).
__global__ void scatter_add(const float* __restrict__ M,
                            const int* __restrict__ src,
                            const int* __restrict__ dst,
                            float* __restrict__ agg, int n_edges) {
  int e = blockIdx.x * (blockDim.x >> 5) + (threadIdx.x >> 5);
  if (e >= n_edges) return;
  int lane = threadIdx.x & 31;
  int s = src[e];
  int d = dst[e];
  const float4 v = *(const float4*)(M + (size_t)s * D + lane * 4);
  float* a = agg + (size_t)d * D + lane * 4;
  atomicAdd(a + 0, v.x);
  atomicAdd(a + 1, v.y);
  atomicAdd(a + 2, v.z);
  atomicAdd(a + 3, v.w);
}

// ---------------- fused SAGE GEMM ----------------
// out[row, n] = (X @ W)[row, n] (+ bias[n]) (+ agg[row,n]*inv_deg[row]) (relu?)
// Block = 256 threads = 8 waves. Block computes rows [16*bid, 16*bid+16),
// each wave owns one 16x16 column tile (8 tiles x 16 = 128 cols).
// K=128 reduced with 32 x V_WMMA_F32_16X16X4_F32.
__launch_bounds__(256, 4)
__global__ void sage_gemm(const float* __restrict__ X, const float* __restrict__ W,
                          const float* __restrict__ bias, const float* __restrict__ agg,
                          const float* __restrict__ inv_deg, float* __restrict__ out,
                          int do_relu) {
  __shared__ float Xs[16 * 132];   // +4 pad: stride 132 breaks 64-bank conflicts

  const int row0 = blockIdx.x * 16;

  // Cooperative load of the 16x128 A tile (8 floats per thread).
  {
    int r  = threadIdx.x >> 4;         // 0..15
    int c0 = (threadIdx.x & 15) * 8;   // 0..120
    const float* xp = X + (size_t)(row0 + r) * D + c0;
    float4 p0 = *(const float4*)xp;
    float4 p1 = *(const float4*)(xp + 4);
    *(float4*)&Xs[r * 132 + c0]     = p0;
    *(float4*)&Xs[r * 132 + c0 + 4] = p1;
  }
  __syncthreads();

  const int wave = threadIdx.x >> 5;       // column tile 0..7
  const int lane = threadIdx.x & 31;
  const int m    = lane & 15;              // A row within tile
  const int kk   = (lane >> 4) << 1;       // A/B k-pair select (0 or 2)
  const int n    = (wave << 4) + (lane & 15);  // global output column

  const float* wp = W + n;                 // column n of W (row-major DxD)

  v8f acc = {};
#pragma unroll
  for (int k = 0; k < D; k += 4) {
    // A frag: lane holds A[m][k+kk], A[m][k+kk+1]
    v2f a = *(const v2f*)&Xs[m * 132 + k + kk];
    // B frag: lane holds W[k+kk][n], W[k+kk+1][n]
    v2f b;
    b.x = wp[(size_t)(k + kk) * D];
    b.y = wp[(size_t)(k + kk + 1) * D];
    acc = __builtin_amdgcn_wmma_f32_16x16x4_f32(
        /*neg_a=*/false, a, /*neg_b=*/false, b,
        /*c_mod=*/(short)0, acc, /*reuse_a=*/false, /*reuse_b=*/false);
  }

  // Epilogue: C/D layout -> VGPR r holds rows r (lanes 0-15) / r+8 (lanes 16-31).
#pragma unroll
  for (int r = 0; r < 8; ++r) {
    int row = row0 + ((lane < 16) ? r : (r + 8));
    float v = acc[r];
    if (bias) v += bias[n];
    if (agg)  v += agg[(size_t)row * D + n] * inv_deg[row];
    if (do_relu) v = fmaxf(v, 0.0f);
    out[(size_t)row * D + n] = v;
  }
}

// ---------------- launch ----------------

extern "C" void kernel_launch(void* const* d_in, const int* in_sizes, int n_in,
                              void* d_out, int out_size, void* d_ws, size_t ws_size,
                              hipStream_t stream) {
  (void)in_sizes; (void)n_in; (void)out_size; (void)ws_size;

  const float* x   = (const float*)d_in[0];
  const int*   ei  = (const int*)d_in[1];
  const float* W0l = (const float*)d_in[2];
  const float* b0l = (const float*)d_in[3];
  const float* W0r = (const float*)d_in[4];
  const float* W1l = (const float*)d_in[5];
  const float* b1l = (const float*)d_in[6];
  const float* W1r = (const float*)d_in[7];
  float* out = (float*)d_out;

  const int* src = ei;             // edge_index[0]
  const int* dst = ei + N_EDGES;   // edge_index[1]

  float* ws      = (float*)d_ws;
  float* inv_deg = ws;                               // 50176 floats (padded)
  float* Mbuf    = ws + 50176;                       // 50000*128 messages
  float* aggb    = Mbuf + (size_t)N_NODES * D;       // 50000*128 aggregate
  float* hbuf    = aggb + (size_t)N_NODES * D;       // 50000*128 hidden

  const long FEAT = (long)N_NODES * D;

  // in-degree -> inverse degree (in place)
  zero_f32<<<256, 256, 0, stream>>>(inv_deg, N_NODES);
  deg_count<<<(N_EDGES + 255) / 256, 256, 0, stream>>>(dst, inv_deg, N_EDGES);
  finalize_inv_deg<<<(N_NODES + 255) / 256, 256, 0, stream>>>(inv_deg, N_NODES);

  // ---- layer 0 ----
  sage_gemm<<<ROW_TILES, 256, 0, stream>>>(x, W0l, b0l, nullptr, inv_deg, Mbuf, 0);
  zero_f32<<<2048, 256, 0, stream>>>(aggb, FEAT);
  scatter_add<<<N_EDGES / 8, 256, 0, stream>>>(Mbuf, src, dst, aggb, N_EDGES);
  sage_gemm<<<ROW_TILES, 256, 0, stream>>>(x, W0r, nullptr, aggb, inv_deg, hbuf, 1);

  // ---- layer 1 ----
  sage_gemm<<<ROW_TILES, 256, 0, stream>>>(hbuf, W1l, b1l, nullptr, inv_deg, Mbuf, 0);
  zero_f32<<<2048, 256, 0, stream>>>(aggb, FEAT);
  scatter_add<<<N_EDGES / 8, 256, 0, stream>>>(Mbuf, src, dst, aggb, N_EDGES);
  sage_gemm<<<ROW_TILES, 256, 0, stream>>>(hbuf, W1r, nullptr, aggb, inv_deg, out, 0);
}